// CBAMLoss_76089640616111
// MI455X (gfx1250) — compile-verified
//
#include <hip/hip_runtime.h>
#include <math.h>

// ---- problem constants (match reference) ----
#define C_CLASSES 1000
#define ALPHA_C   0.5f
#define BETA_C    0.3f
#define MAGIC_C   0.165745444183859f
#define E1_C      60
#define E2_C      80

typedef __attribute__((ext_vector_type(2))) float v2f;
typedef __attribute__((ext_vector_type(8))) float v8f;

__device__ __forceinline__ float wave_sum(float v) {
#pragma unroll
    for (int m = 16; m >= 1; m >>= 1) v += __shfl_xor(v, m, 32);
    return v;
}
__device__ __forceinline__ float wave_max(float v) {
#pragma unroll
    for (int m = 16; m >= 1; m >>= 1) v = fmaxf(v, __shfl_xor(v, m, 32));
    return v;
}
__device__ __forceinline__ float wave_min(float v) {
#pragma unroll
    for (int m = 16; m >= 1; m >>= 1) v = fminf(v, __shfl_xor(v, m, 32));
    return v;
}

// ---------------------------------------------------------------------------
// Kernel 0: per-class margin precompute (C=1000, one block of 1024 threads)
// m_list1[c] = max_m * (1/sqrt(n_c)) / (1/sqrt(n_min))
//            + (ALPHA*d_c^2+BETA) * (max_m / max_w) * ee / 2
// ---------------------------------------------------------------------------
__global__ __launch_bounds__(1024) void margins_kernel(
    const float* __restrict__ cls_num, const float* __restrict__ diff,
    const int* __restrict__ epoch_p, float* __restrict__ m1_out)
{
    __shared__ float s_min[32], s_sum[32], s_max[32];
    __shared__ float g_min, g_sum, g_max;

    const int tid  = threadIdx.x;
    const int lane = tid & 31;
    const int wid  = tid >> 5;

    float n = 0.0f, d = 0.0f;
    float vmin = INFINITY, vsum = 0.0f, vmax = 0.0f;  // wraw >= 0.3, so 0 is a safe pad
    if (tid < C_CLASSES) {
        n = cls_num[tid];
        d = diff[tid];
        vmin = n;
        vsum = n;
        vmax = ALPHA_C * d * d + BETA_C;
    }
    vmin = wave_min(vmin);
    vsum = wave_sum(vsum);
    vmax = wave_max(vmax);
    if (lane == 0) { s_min[wid] = vmin; s_sum[wid] = vsum; s_max[wid] = vmax; }
    __syncthreads();
    if (wid == 0) {
        float a = wave_min(s_min[lane]);
        float b = wave_sum(s_sum[lane]);
        float c = wave_max(s_max[lane]);
        if (lane == 0) { g_min = a; g_sum = b; g_max = c; }
    }
    __syncthreads();

    if (tid < C_CLASSES) {
        const float nmin = g_min, nsum = g_sum, wmax = g_max;
        const float max_m = -logf(nmin / nsum) - MAGIC_C;   // mirrors reference
        const float maxp  = 1.0f / sqrtf(nmin);             // max of cls_p
        const int   epoch = epoch_p[0];
        float ee;
        if (epoch < E1_C)       ee = 0.0f;
        else if (epoch >= E2_C) ee = 1.0f;
        else                    ee = (float)(epoch - E1_C) / (float)(E2_C - E1_C);

        const float clsp = 1.0f / sqrtf(n);
        const float m    = max_m * clsp / maxp;
        const float w    = (ALPHA_C * d * d + BETA_C) * (max_m / wmax);
        m1_out[tid] = m + w * ee * 0.5f;                    // + weights*ee/2
    }
}

// ---------------------------------------------------------------------------
// Kernel 1: one row (1000 f32) per wave32. Entire row lives in 8 float4 regs
// per lane -> single memory pass: max, then sum(exp(x-max)), margin applied
// at the target column in-register. Writes per-row loss to workspace.
// ---------------------------------------------------------------------------
__global__ __launch_bounds__(256) void rowloss_kernel(
    const float* __restrict__ x, const int* __restrict__ targets,
    const float* __restrict__ m1, float* __restrict__ rowloss, int Brows)
{
    const int lane = threadIdx.x & 31;
    const int row  = blockIdx.x * (blockDim.x >> 5) + (threadIdx.x >> 5);
    if (row >= Brows) return;

    const float* __restrict__ xr = x + (size_t)row * C_CLASSES;  // 4000B rows: 16B aligned
    const int   t  = targets[row];
    const float bm = m1[t];

    float4 v[8];
#pragma unroll
    for (int it = 0; it < 8; ++it) {
        const int col = it * 128 + lane * 4;
        if (col < C_CLASSES) {       // C=1000 is a multiple of 4: float4 fully in/out
            v[it] = *reinterpret_cast<const float4*>(xr + col);
        } else {
            v[it] = make_float4(-INFINITY, -INFINITY, -INFINITY, -INFINITY);
        }
    }

    // subtract margin at the target logit (static indexing, no spills)
#pragma unroll
    for (int it = 0; it < 8; ++it) {
        const int col = it * 128 + lane * 4;
        if (col + 0 == t) v[it].x -= bm;
        if (col + 1 == t) v[it].y -= bm;
        if (col + 2 == t) v[it].z -= bm;
        if (col + 3 == t) v[it].w -= bm;
    }

    // row max
    float pm = -INFINITY;
#pragma unroll
    for (int it = 0; it < 8; ++it) {
        pm = fmaxf(pm, fmaxf(fmaxf(v[it].x, v[it].y), fmaxf(v[it].z, v[it].w)));
    }
    const float M = wave_max(pm);

    // sum of exp(x - M); exp(-inf - M) == 0 handles tail pads
    float ps = 0.0f;
#pragma unroll
    for (int it = 0; it < 8; ++it) {
        ps += expf(v[it].x - M);
        ps += expf(v[it].y - M);
        ps += expf(v[it].z - M);
        ps += expf(v[it].w - M);
    }
    const float S = wave_sum(ps);

    if (lane == 0) {
        const float xt = xr[t] - bm;                 // modified target logit
        rowloss[row] = logf(S) + M - xt;             // -log_softmax at target
    }
}

// ---------------------------------------------------------------------------
// Kernel 2: deterministic mean over B row losses. Final 32 wave-partials are
// collapsed with one V_WMMA_F32_16X16X4_F32 against an all-ones B matrix:
// with B==1s, D[i][j] = sum_k A[i][k]; column N=0 of D (lanes 0 and 16)
// yields the grand total after one shfl_xor(16).
// ---------------------------------------------------------------------------
__global__ __launch_bounds__(1024) void reduce_kernel(
    const float* __restrict__ rowloss, float* __restrict__ out, int n)
{
    __shared__ float s_part[32];
    const int tid  = threadIdx.x;
    const int lane = tid & 31;
    const int wid  = tid >> 5;

    float s = 0.0f;
    for (int i = tid; i < n; i += 1024) s += rowloss[i];   // fixed order
    s = wave_sum(s);
    if (lane == 0) s_part[wid] = s;
    __syncthreads();

    if (wid == 0) {
        const float p = s_part[lane];                      // 32 partials
        // A (16x4 f32, 2 VGPRs/lane): lanes 0-15 -> (M=lane, K=0), lanes 16-31 -> (M=lane-16, K=2)
        v2f a; a[0] = p;    a[1] = 0.0f;
        v2f b; b[0] = 1.0f; b[1] = 1.0f;                   // all-ones B => row sums
        v8f c = {};
        v8f d2 = __builtin_amdgcn_wmma_f32_16x16x4_f32(
            /*neg_a=*/false, a, /*neg_b=*/false, b,
            /*c_mod=*/(short)0, c, /*reuse_a=*/false, /*reuse_b=*/false);
        // lane 0 holds D[M=0..7][N=0] in d2[0..7]; lane 16 holds D[M=8..15][N=0]
        float tsum = d2[0] + d2[1] + d2[2] + d2[3] + d2[4] + d2[5] + d2[6] + d2[7];
        tsum += __shfl_xor(tsum, 16, 32);
        if (lane == 0) out[0] = tsum / (float)n;
    }
}

// ---------------------------------------------------------------------------
extern "C" void kernel_launch(void* const* d_in, const int* in_sizes, int n_in,
                              void* d_out, int out_size, void* d_ws, size_t ws_size,
                              hipStream_t stream)
{
    const float* x        = (const float*)d_in[0];
    const int*   targets  = (const int*)  d_in[1];
    const float* cls_num  = (const float*)d_in[2];
    const float* diff     = (const float*)d_in[3];
    const int*   epoch    = (const int*)  d_in[4];

    const int Brows = in_sizes[1];          // = 32768

    float* m1      = (float*)d_ws;          // [0 .. 1023]   per-class margins (1000 used)
    float* rowloss = m1 + 1024;             // [1024 .. 1024+B) per-row losses

    margins_kernel<<<1, 1024, 0, stream>>>(cls_num, diff, epoch, m1);

    const int rowsPerBlock = 256 / 32;      // 8 waves -> 8 rows per block
    const int nBlocks = (Brows + rowsPerBlock - 1) / rowsPerBlock;
    rowloss_kernel<<<nBlocks, 256, 0, stream>>>(x, targets, m1, rowloss, Brows);

    reduce_kernel<<<1, 1024, 0, stream>>>(rowloss, (float*)d_out, Brows);
}